// MultiHeadGATODE_57655640981627
// MI455X (gfx1250) — compile-verified
//
#include <hip/hip_runtime.h>

typedef __attribute__((ext_vector_type(2))) float v2f;
typedef __attribute__((ext_vector_type(8))) float v8f;

#define IN_DIM  512          // input feature dim
#define OUT_TOT 512          // H * OUT = 4 * 128 (concatenated heads)

// ---------------------------------------------------------------------------
// Stage 1: fused all-heads GEMM  x_all = (h @ W_all^T + b) * norm[row]
//   grid.x = row tiles (16 rows each), block = 256 threads = 8 waves.
//   Wave w computes a 16x64 output tile (cols w*64 .. w*64+63) as 4 WMMA accs,
//   so the whole block covers a full 16 x 512 output stripe.
//   A fragment (f32 16x4): lane L holds row (L&15), K pair at k + 2*(L>>4).
//   B fragment (f32 4x16): lane L holds col (L&15), same K-pair striping.
// ---------------------------------------------------------------------------
__global__ __launch_bounds__(256) void gat_gemm_wmma(
    const float* __restrict__ h,     // [N, 512]
    const float* __restrict__ W,     // [512, 512] (= [H,OUT,IN] flattened)
    const float* __restrict__ bias,  // [512]
    const float* __restrict__ norm,  // [N]
    float* __restrict__ xall,        // [N, 512] workspace
    int N)
{
    const int lane  = threadIdx.x & 31;
    const int wave  = threadIdx.x >> 5;       // 0..7
    const int half  = lane >> 4;              // 0 or 1
    const int idx16 = lane & 15;
    const int kb    = half << 1;              // 0 or 2: K sub-pair per half-wave

    const int row_tile = blockIdx.x;
    int row  = row_tile * 16 + idx16;
    int arow = row < N ? row : N - 1;         // clamp load address (uniform flow)
    const float* hrow = h + (size_t)arow * IN_DIM;

    const int col0 = wave * 64 + idx16;
    const float* wp0 = W + (size_t)(col0)      * IN_DIM;
    const float* wp1 = W + (size_t)(col0 + 16) * IN_DIM;
    const float* wp2 = W + (size_t)(col0 + 32) * IN_DIM;
    const float* wp3 = W + (size_t)(col0 + 48) * IN_DIM;

    v8f acc0 = {}, acc1 = {}, acc2 = {}, acc3 = {};

#pragma unroll 4
    for (int k = 0; k < IN_DIM; k += 4) {
        v2f a  = *(const v2f*)(hrow + k + kb);
        v2f b0 = *(const v2f*)(wp0  + k + kb);
        v2f b1 = *(const v2f*)(wp1  + k + kb);
        v2f b2 = *(const v2f*)(wp2  + k + kb);
        v2f b3 = *(const v2f*)(wp3  + k + kb);
        acc0 = __builtin_amdgcn_wmma_f32_16x16x4_f32(false, a, false, b0,
                                                     (short)0, acc0, false, false);
        acc1 = __builtin_amdgcn_wmma_f32_16x16x4_f32(false, a, false, b1,
                                                     (short)0, acc1, false, false);
        acc2 = __builtin_amdgcn_wmma_f32_16x16x4_f32(false, a, false, b2,
                                                     (short)0, acc2, false, false);
        acc3 = __builtin_amdgcn_wmma_f32_16x16x4_f32(false, a, false, b3,
                                                     (short)0, acc3, false, false);
    }

    // C/D layout: VGPR v holds row (v + 8*half) of the tile, col = lane&15.
    const int rbase = row_tile * 16 + half * 8;

    // norm[rbase .. rbase+7]: consecutive -> two b128 loads on the full-tile
    // (common) path; clamped scalar loads only for a ragged last tile.
    float nrm[8];
    if (rbase + 7 < N) {                       // uniform branch per half-wave
        float4 n0 = *(const float4*)(norm + rbase);
        float4 n1 = *(const float4*)(norm + rbase + 4);
        nrm[0] = n0.x; nrm[1] = n0.y; nrm[2] = n0.z; nrm[3] = n0.w;
        nrm[4] = n1.x; nrm[5] = n1.y; nrm[6] = n1.z; nrm[7] = n1.w;
    } else {
#pragma unroll
        for (int v = 0; v < 8; ++v) {
            int r = rbase + v;
            nrm[v] = norm[r < N ? r : N - 1];
        }
    }

    const v8f accs[4] = {acc0, acc1, acc2, acc3};
#pragma unroll
    for (int t = 0; t < 4; ++t) {
        int col  = wave * 64 + t * 16 + idx16;
        float bv = bias[col];
        v8f a = accs[t];
#pragma unroll
        for (int v = 0; v < 8; ++v) {
            int r = rbase + v;
            if (r < N)
                xall[(size_t)r * OUT_TOT + col] = (a[v] + bv) * nrm[v];
        }
    }
}

// ---------------------------------------------------------------------------
// Stage 2: edge scatter-add.  ONE WAVE PER EDGE: the edge index is
// wave-uniform, so src/dst are pinned to SGPRs (readfirstlane) and loaded
// once per wave instead of once per thread.  Each lane moves 16 floats:
// 4x float4 loads from x_all[src] + 16 no-return global_atomic_add_f32.
// ---------------------------------------------------------------------------
__global__ __launch_bounds__(256) void gat_scatter(
    const float* __restrict__ xall,
    const int* __restrict__ src,
    const int* __restrict__ dst,
    float* __restrict__ out,
    int E)
{
    int e = blockIdx.x * 8 + (threadIdx.x >> 5);   // 8 waves per block
    if (e >= E) return;
    e = __builtin_amdgcn_readfirstlane(e);          // wave-uniform edge id
    const int lane = threadIdx.x & 31;

    int s = __builtin_amdgcn_readfirstlane(src[e]);
    int d = __builtin_amdgcn_readfirstlane(dst[e]);

    const float* xs = xall + (size_t)s * OUT_TOT;
    float*       od = out  + (size_t)d * OUT_TOT;

#pragma unroll
    for (int i = 0; i < 4; ++i) {
        int f = i * 128 + lane * 4;                 // 32 lanes x float4 = 128
        const float4 v = *(const float4*)(xs + f);
        atomicAdd(od + f + 0, v.x);
        atomicAdd(od + f + 1, v.y);
        atomicAdd(od + f + 2, v.z);
        atomicAdd(od + f + 3, v.w);
    }
}

// ---------------------------------------------------------------------------
// Stage 3: out = relu(out * norm[row]), float4 per thread.
// ---------------------------------------------------------------------------
__global__ __launch_bounds__(256) void gat_finalize(
    float* __restrict__ out,
    const float* __restrict__ norm,
    int total4)   // N * 512 / 4
{
    int idx = blockIdx.x * blockDim.x + threadIdx.x;
    if (idx >= total4) return;
    int n = idx >> 7;                          // (idx*4) / 512
    float nv = norm[n];
    float4 v = *((const float4*)out + idx);
    v.x = fmaxf(v.x * nv, 0.0f);
    v.y = fmaxf(v.y * nv, 0.0f);
    v.z = fmaxf(v.z * nv, 0.0f);
    v.w = fmaxf(v.w * nv, 0.0f);
    *((float4*)out + idx) = v;
}

extern "C" void kernel_launch(void* const* d_in, const int* in_sizes, int n_in,
                              void* d_out, int out_size, void* d_ws, size_t ws_size,
                              hipStream_t stream)
{
    const float* h    = (const float*)d_in[0];   // [N, 512]
    const float* W    = (const float*)d_in[1];   // [4, 128, 512]
    const float* bias = (const float*)d_in[2];   // [4, 128]
    const float* norm = (const float*)d_in[3];   // [N]
    const int*   src  = (const int*)d_in[4];     // [E]
    const int*   dst  = (const int*)d_in[5];     // [E]

    const int N = in_sizes[3];                   // norm has N elements
    const int E = in_sizes[4];

    float* out  = (float*)d_out;
    float* xall = (float*)d_ws;                  // [N, 512] scratch

    // zero the accumulation target (graph-capturable memset node)
    hipMemsetAsync(d_out, 0, (size_t)out_size * sizeof(float), stream);

    // Stage 1: WMMA GEMM + bias + pre-norm scale
    int row_tiles = (N + 15) / 16;
    gat_gemm_wmma<<<row_tiles, 256, 0, stream>>>(h, W, bias, norm, xall, N);

    // Stage 2: scatter-add over edges, one wave per edge
    int sc_blocks = (E + 7) / 8;
    gat_scatter<<<sc_blocks, 256, 0, stream>>>(xall, src, dst, out, E);

    // Stage 3: post-norm scale + relu
    int total4 = out_size / 4;
    int fin_blocks = (total4 + 255) / 256;
    gat_finalize<<<fin_blocks, 256, 0, stream>>>(out, norm, total4);
}